// CvtLstm_60730837565574
// MI455X (gfx1250) — compile-verified
//
#include <hip/hip_runtime.h>
#include <math.h>

// ---------------------------------------------------------------------------
// CvtLstm (first step, h=c=0) for gfx1250 / MI455X.
// All GEMMs run on V_WMMA_F32_16X16X4_F32 (exact fp32 matrix math).
// One wave (wave32) computes a 64x16 output tile (4 stacked 16x16 WMMA tiles)
// so B-fragment loads and im2col index math are amortized over 4 WMMAs.
// Fragments follow the CDNA5 ISA 7.12.2 VGPR layouts for 32-bit A/B/C.
// ---------------------------------------------------------------------------

typedef __attribute__((ext_vector_type(2))) float v2f;
typedef __attribute__((ext_vector_type(8))) float v8f;

#define NB   8      // batch
#define ICH  128    // input channels
#define RCH  256    // recurrent / attention / output channels
#define HWQ  1296   // 36*36 query tokens per image
#define DKV  1156   // 34*34 key/value tokens per image
#define K33  2304   // 256*9 im2col depth
#define TILQ 81     // 1296/16 token tiles per image (exact)
#define TILV 73     // ceil(1156/16) token tiles per image (last partial)
#define MT   4      // 16-row M tiles per wave

__device__ __forceinline__ v8f wmma4(float a0, float a1, float b0, float b1, v8f c) {
  v2f a = {a0, a1};
  v2f b = {b0, b1};
  // 8-arg form: (neg_a, A, neg_b, B, c_mod, C, reuse_a, reuse_b)
  return __builtin_amdgcn_wmma_f32_16x16x4_f32(false, a, false, b, (short)0, c,
                                               false, false);
}

__device__ __forceinline__ float hmax16(float v) {
  // reduce across the 16 lanes of each wave half (xor masks stay within half)
  #pragma unroll
  for (int m = 1; m < 16; m <<= 1) v = fmaxf(v, __shfl_xor(v, m, 32));
  return v;
}
__device__ __forceinline__ float hsum16(float v) {
  #pragma unroll
  for (int m = 1; m < 16; m <<= 1) v += __shfl_xor(v, m, 32);
  return v;
}

// ---------------------------------------------------------------------------
// Generic 1x1-conv GEMM: Out[n][m][hw] = bias[m] + sum_k W1[m][k]*B1[n][k][hw]
//                                       (+ sum_k W2[m][k]*B2[n][k][hw])
// Token geometry fixed to 36x36 (1296 = 81 tiles per image, exact).
// Each wave: 64 output rows x 16 tokens. M must be a multiple of 64.
// ---------------------------------------------------------------------------
__global__ void gemm1x1_wmma(const float* __restrict__ W1, const float* __restrict__ B1,
                             const float* __restrict__ W2, const float* __restrict__ B2,
                             const float* __restrict__ bias, float* __restrict__ Out,
                             int M, int K) {
  const int tilesT = NB * TILQ;  // 648
  int wv = blockIdx.x * (blockDim.x >> 5) + (threadIdx.x >> 5);
  int total = (M >> 6) * tilesT;
  if (wv >= total) return;
  int tileG = wv / tilesT;
  int tt = wv - tileG * tilesT;
  int n = tt / TILQ;
  int hw0 = (tt - n * TILQ) << 4;
  int lane = threadIdx.x & 31, half = lane >> 4, lm = lane & 15;
  int m0 = tileG << 6;  // 64 rows per wave
  int hw = hw0 + lm;

  v8f acc[MT] = {};
  if (bias) {
    #pragma unroll
    for (int mt = 0; mt < MT; ++mt)
      #pragma unroll
      for (int r = 0; r < 8; ++r) acc[mt][r] = bias[m0 + 16 * mt + r + 8 * half];
  }
  {
    const float* wr = W1 + (size_t)(m0 + lm) * K;
    const float* br = B1 + (size_t)n * K * HWQ + hw;
    for (int k0 = 0; k0 < K; k0 += 4) {
      int ka = k0 + 2 * half;
      float b0 = br[(size_t)ka * HWQ];
      float b1 = br[(size_t)(ka + 1) * HWQ];
      #pragma unroll
      for (int mt = 0; mt < MT; ++mt) {
        float a0 = wr[(size_t)(16 * mt) * K + ka];
        float a1 = wr[(size_t)(16 * mt) * K + ka + 1];
        acc[mt] = wmma4(a0, a1, b0, b1, acc[mt]);
      }
    }
  }
  if (W2) {
    const float* wr = W2 + (size_t)(m0 + lm) * K;
    const float* br = B2 + (size_t)n * K * HWQ + hw;
    for (int k0 = 0; k0 < K; k0 += 4) {
      int ka = k0 + 2 * half;
      float b0 = br[(size_t)ka * HWQ];
      float b1 = br[(size_t)(ka + 1) * HWQ];
      #pragma unroll
      for (int mt = 0; mt < MT; ++mt) {
        float a0 = wr[(size_t)(16 * mt) * K + ka];
        float a1 = wr[(size_t)(16 * mt) * K + ka + 1];
        acc[mt] = wmma4(a0, a1, b0, b1, acc[mt]);
      }
    }
  }
  size_t outb = (size_t)n * M * HWQ;
  #pragma unroll
  for (int mt = 0; mt < MT; ++mt)
    #pragma unroll
    for (int r = 0; r < 8; ++r)
      Out[outb + (size_t)(m0 + 16 * mt + r + 8 * half) * HWQ + hw] = acc[mt][r];
}

// ---------------------------------------------------------------------------
// 3x3 conv as im2col GEMM over K=2304. same=1: 36x36 SAME; same=0: 34x34 VALID.
// Loads are address-clamped + value-masked (branchless) so EXEC stays full
// for WMMA. Each wave: 64 output channels x 16 tokens; the im2col gather and
// its index math are shared by 4 WMMAs.
// ---------------------------------------------------------------------------
__device__ __forceinline__ float im2col_load(const float* __restrict__ xb, int k,
                                             int y, int x, int pad) {
  int c = k / 9;
  int r9 = k - c * 9;
  int dy = r9 / 3;
  int dx = r9 - dy * 3;
  int iy = y + dy - pad;
  int ix = x + dx - pad;
  bool ok = ((unsigned)iy < 36u) & ((unsigned)ix < 36u);
  int cy = ok ? iy : 0;
  int cx = ok ? ix : 0;
  float v = xb[(size_t)c * HWQ + cy * 36 + cx];
  return ok ? v : 0.0f;
}

__global__ void conv3x3_wmma(const float* __restrict__ Wt, const float* __restrict__ Xin,
                             float* __restrict__ Out, int same) {
  const int outW = same ? 36 : 34;
  const int outHW = outW * outW;
  const int tpi = same ? TILQ : TILV;
  const int tilesT = NB * tpi;
  int wv = blockIdx.x * (blockDim.x >> 5) + (threadIdx.x >> 5);
  int total = (RCH >> 6) * tilesT;  // 4 M-groups of 64 rows
  if (wv >= total) return;
  int tileG = wv / tilesT;
  int tt = wv - tileG * tilesT;
  int n = tt / tpi;
  int hw0 = (tt - n * tpi) << 4;
  int lane = threadIdx.x & 31, half = lane >> 4, lm = lane & 15;
  int hw = hw0 + lm;
  bool tok = hw < outHW;
  int hwc = tok ? hw : 0;
  int y = hwc / outW;
  int x = hwc - y * outW;
  int pad = same ? 1 : 0;
  int m0 = tileG << 6;
  const float* xb = Xin + (size_t)n * RCH * HWQ;
  const float* wr = Wt + (size_t)(m0 + lm) * K33;

  v8f acc[MT] = {};
  for (int k0 = 0; k0 < K33; k0 += 4) {
    int ka = k0 + 2 * half;
    float b0 = im2col_load(xb, ka, y, x, pad);
    float b1 = im2col_load(xb, ka + 1, y, x, pad);
    #pragma unroll
    for (int mt = 0; mt < MT; ++mt) {
      float a0 = wr[(size_t)(16 * mt) * K33 + ka];
      float a1 = wr[(size_t)(16 * mt) * K33 + ka + 1];
      acc[mt] = wmma4(a0, a1, b0, b1, acc[mt]);
    }
  }
  if (tok) {
    size_t outb = (size_t)n * RCH * outHW;
    #pragma unroll
    for (int mt = 0; mt < MT; ++mt)
      #pragma unroll
      for (int r = 0; r < 8; ++r)
        Out[outb + (size_t)(m0 + 16 * mt + r + 8 * half) * outHW + hw] = acc[mt][r];
  }
}

// ---------------------------------------------------------------------------
// Flash-style attention per (n, head, 16-query tile). Q/K/V stored channel-
// major: [N][256][tokens], per head a 32x{1296|1156} slice. Online softmax,
// P transposed through a per-wave 1KB LDS slab (in-wave DS ordering, no
// barrier needed).
// ---------------------------------------------------------------------------
__global__ void attn_wmma(const float* __restrict__ Q, const float* __restrict__ Kt,
                          const float* __restrict__ V, float* __restrict__ Aout) {
  __shared__ float pshare[8 * 256];  // 8 waves/block x 16x16 P tile
  int wv = blockIdx.x * (blockDim.x >> 5) + (threadIdx.x >> 5);
  const int total = NB * 8 * TILQ;  // 5184
  if (wv >= total) return;
  int qt = wv % TILQ;
  int nh = wv / TILQ;
  int n = nh >> 3;
  int head = nh & 7;
  int lane = threadIdx.x & 31, half = lane >> 4, lm = lane & 15;
  float* pb = pshare + ((threadIdx.x >> 5) << 8);
  const size_t qoff = (size_t)n * RCH * HWQ + (size_t)head * 32 * HWQ;
  const size_t koff = (size_t)n * RCH * DKV + (size_t)head * 32 * DKV;
  int q0 = qt << 4;

  // Q A-fragments for all 8 K-steps (K=32 channels), kept resident.
  float qa0[8], qa1[8];
  #pragma unroll
  for (int s = 0; s < 8; ++s) {
    int c = 4 * s + 2 * half;
    qa0[s] = Q[qoff + (size_t)c * HWQ + q0 + lm];
    qa1[s] = Q[qoff + (size_t)(c + 1) * HWQ + q0 + lm];
  }

  float mr[8], lr[8];
  v8f o0 = {}, o1 = {};
  #pragma unroll
  for (int r = 0; r < 8; ++r) { mr[r] = -INFINITY; lr[r] = 0.0f; }

  for (int kt = 0; kt < TILV; ++kt) {
    int kv0 = kt << 4;
    int col = kv0 + lm;
    bool cok = col < DKV;
    int colc = cok ? col : (DKV - 1);

    // S tile = Q^T K  (M=q rows, N=kv cols, K=32)
    v8f s = {};
    #pragma unroll
    for (int st = 0; st < 8; ++st) {
      int c = 4 * st + 2 * half;
      float b0 = Kt[koff + (size_t)c * DKV + colc];
      float b1 = Kt[koff + (size_t)(c + 1) * DKV + colc];
      s = wmma4(qa0[st], qa1[st], b0, b1, s);
    }
    #pragma unroll
    for (int r = 0; r < 8; ++r) s[r] = cok ? s[r] : -INFINITY;

    // Online softmax update + stash P (D-layout -> [q][kv] in LDS).
    #pragma unroll
    for (int r = 0; r < 8; ++r) {
      float tm = hmax16(s[r]);
      float nm = fmaxf(mr[r], tm);
      float sc = __expf(mr[r] - nm);  // first tile: exp(-inf)=0
      float p = __expf(s[r] - nm);    // masked cols: exp(-inf)=0
      float ts = hsum16(p);
      lr[r] = lr[r] * sc + ts;
      mr[r] = nm;
      o0[r] *= sc;
      o1[r] *= sc;
      pb[(r + 8 * half) * 16 + lm] = p;
    }

    // O += P * V^T  (M=q, N=c (two 16-wide tiles), K=16 kv)
    #pragma unroll
    for (int t4 = 0; t4 < 4; ++t4) {
      float pa0 = pb[lm * 16 + 4 * t4 + 2 * half];
      float pa1 = pb[lm * 16 + 4 * t4 + 2 * half + 1];
      int kvr = kv0 + 4 * t4 + 2 * half;
      int kva = (kvr < DKV) ? kvr : (DKV - 1);
      int kvb = (kvr + 1 < DKV) ? (kvr + 1) : (DKV - 1);
      float vb0 = V[koff + (size_t)lm * DKV + kva];
      float vb1 = V[koff + (size_t)lm * DKV + kvb];
      o0 = wmma4(pa0, pa1, vb0, vb1, o0);
      float vc0 = V[koff + (size_t)(16 + lm) * DKV + kva];
      float vc1 = V[koff + (size_t)(16 + lm) * DKV + kvb];
      o1 = wmma4(pa0, pa1, vc0, vc1, o1);
    }
  }

  // a[n][head*32+c][q] = O[q][c] / l[q]
  #pragma unroll
  for (int r = 0; r < 8; ++r) {
    float inv = 1.0f / lr[r];
    int qrow = q0 + r + 8 * half;
    Aout[qoff + (size_t)lm * HWQ + qrow] = o0[r] * inv;
    Aout[qoff + (size_t)(16 + lm) * HWQ + qrow] = o1[r] * inv;
  }
}

// ---------------------------------------------------------------------------
// LSTM pointwise with c_prev = 0: c = sig(gi)*tanh(gg); hn = sig(go)*tanh(c).
// (forget gate multiplies zero state -> dropped exactly)
// ---------------------------------------------------------------------------
__global__ void lstm_pointwise(const float* __restrict__ G, float* __restrict__ Hn) {
  int idx = blockIdx.x * blockDim.x + threadIdx.x;
  const int tot = NB * RCH * HWQ;
  if (idx >= tot) return;
  int n = idx / (RCH * HWQ);
  int rem = idx - n * RCH * HWQ;
  int m = rem / HWQ;
  int hw = rem - m * HWQ;
  const float* gb = G + (size_t)n * 4 * RCH * HWQ;
  float gi = gb[(size_t)m * HWQ + hw];
  float gg = gb[(size_t)(m + 2 * RCH) * HWQ + hw];
  float go = gb[(size_t)(m + 3 * RCH) * HWQ + hw];
  float si = 1.0f / (1.0f + __expf(-gi));
  float so = 1.0f / (1.0f + __expf(-go));
  float cc = si * tanhf(gg);
  Hn[idx] = so * tanhf(cc);
}

// ---------------------------------------------------------------------------
// Host-side launch. Inputs (setup_inputs order):
// 0:x 1:w_in 2:b_in 3:w_qx 4:w_qh 5:w_kx 6:w_kh 7:w_vx 8:w_vh
// 9:w_ga 10:b_g 11:w_gx 12:w_gh 13:w_out 14:b_out   (all *_h unused: h=0)
// ---------------------------------------------------------------------------
extern "C" void kernel_launch(void* const* d_in, const int* in_sizes, int n_in,
                              void* d_out, int out_size, void* d_ws, size_t ws_size,
                              hipStream_t stream) {
  (void)in_sizes; (void)n_in; (void)out_size; (void)ws_size;
  const float* x     = (const float*)d_in[0];
  const float* w_in  = (const float*)d_in[1];
  const float* b_in  = (const float*)d_in[2];
  const float* w_qx  = (const float*)d_in[3];
  const float* w_kx  = (const float*)d_in[5];
  const float* w_vx  = (const float*)d_in[7];
  const float* w_ga  = (const float*)d_in[9];
  const float* b_g   = (const float*)d_in[10];
  const float* w_gx  = (const float*)d_in[11];
  const float* w_out = (const float*)d_in[13];
  const float* b_out = (const float*)d_in[14];
  float* out = (float*)d_out;

  float* ws = (float*)d_ws;
  size_t off = 0;
  float* xin = ws + off; off += (size_t)NB * RCH * HWQ;       // 2,654,208
  float* qb  = ws + off; off += (size_t)NB * RCH * HWQ;
  float* kb  = ws + off; off += (size_t)NB * RCH * DKV;       // 2,367,488
  float* vb  = ws + off; off += (size_t)NB * RCH * DKV;
  float* ab  = ws + off; off += (size_t)NB * RCH * HWQ;
  float* gb  = ws + off; off += (size_t)NB * 4 * RCH * HWQ;   // 10,616,832
  float* hb  = ws + off; off += (size_t)NB * RCH * HWQ;
  // total ~26.0M floats (~104 MB) of d_ws

  dim3 blk(256);  // 8 waves per block
  auto nblk = [](int waves) { return (unsigned)((waves + 7) / 8); };

  // xin = conv1x1(x, w_in) + b_in           (M=256 -> 4 M-groups)
  gemm1x1_wmma<<<nblk(4 * 648), blk, 0, stream>>>(w_in, x, nullptr, nullptr,
                                                  b_in, xin, RCH, ICH);
  // q / k / v (h=0 drops the *_h convs exactly)
  conv3x3_wmma<<<nblk(4 * NB * TILQ), blk, 0, stream>>>(w_qx, xin, qb, 1);
  conv3x3_wmma<<<nblk(4 * NB * TILV), blk, 0, stream>>>(w_kx, xin, kb, 0);
  conv3x3_wmma<<<nblk(4 * NB * TILV), blk, 0, stream>>>(w_vx, xin, vb, 0);
  // attention -> a
  attn_wmma<<<nblk(NB * 8 * TILQ), blk, 0, stream>>>(qb, kb, vb, ab);
  // gates = w_ga*a + b_g + w_gx*xin  (w_gh*h = 0)   (M=1024 -> 16 M-groups)
  gemm1x1_wmma<<<nblk(16 * 648), blk, 0, stream>>>(w_ga, ab, w_gx, xin,
                                                   b_g, gb, 4 * RCH, RCH);
  // LSTM pointwise -> hn
  lstm_pointwise<<<(NB * RCH * HWQ + 255) / 256, blk, 0, stream>>>(gb, hb);
  // out = conv1x1(hn, w_out) + b_out        (M=256 -> 4 M-groups)
  gemm1x1_wmma<<<nblk(4 * 648), blk, 0, stream>>>(w_out, hb, nullptr, nullptr,
                                                  b_out, out, RCH, RCH);
}